// DGNN_52501680226716
// MI455X (gfx1250) — compile-verified
//
#include <hip/hip_runtime.h>
#include <hip/hip_bf16.h>
#include <math.h>

#define D 64
#define EPS 1e-5f

typedef __attribute__((ext_vector_type(2))) float v2f;
typedef __attribute__((ext_vector_type(8))) float v8f;

// ---- monotone float <-> u32 ordering for atomicMax-based segment max ----
__device__ __forceinline__ unsigned ordKey(float f) {
    unsigned u = __float_as_uint(f);
    return (u & 0x80000000u) ? ~u : (u | 0x80000000u);
}
__device__ __forceinline__ float ordVal(unsigned k) {
    return (k & 0x80000000u) ? __uint_as_float(k & 0x7FFFFFFFu)
                             : __uint_as_float(~k);
}

// ---- generic fill ----
__global__ void fill_f32(float* p, float v, long n) {
    long i = blockIdx.x * (long)blockDim.x + threadIdx.x;
    long stride = (long)gridDim.x * blockDim.x;
    for (; i < n; i += stride) p[i] = v;
}

// ---- edge-weight batch stats (sum, sumsq) ----
__global__ void ew_stats_k(const float* __restrict__ ew, int E, float* stats) {
    __shared__ float ss[256], sq[256];
    float s = 0.f, q = 0.f;
    for (long i = blockIdx.x * (long)blockDim.x + threadIdx.x; i < E;
         i += (long)gridDim.x * blockDim.x) {
        float v = ew[i]; s += v; q += v * v;
    }
    ss[threadIdx.x] = s; sq[threadIdx.x] = q; __syncthreads();
    for (int o = 128; o > 0; o >>= 1) {
        if ((int)threadIdx.x < o) {
            ss[threadIdx.x] += ss[threadIdx.x + o];
            sq[threadIdx.x] += sq[threadIdx.x + o];
        }
        __syncthreads();
    }
    if (threadIdx.x == 0) { atomicAdd(&stats[0], ss[0]); atomicAdd(&stats[1], sq[0]); }
}
__global__ void ew_final_k(float* stats, int E) {
    float mean = stats[0] / (float)E;
    float var  = stats[1] / (float)E - mean * mean;
    stats[2] = mean; stats[3] = var;
}

// ---- segment max of s = -delta*(nt[dst]-et) per destination node ----
__global__ void edge_max_k(const int* __restrict__ dst, const float* __restrict__ et,
                           const float* __restrict__ nt, const float* __restrict__ delta,
                           unsigned* smax, int E) {
    int e = blockIdx.x * blockDim.x + threadIdx.x;
    if (e >= E) return;
    int d = dst[e];
    float td = nt[d] - et[e];
    float s = -delta[0] * td;
    atomicMax(&smax[d], ordKey(s));
}

// ---- softmax denominator + (optional) pos scatter-sum ----
__global__ void edge_den_k(const int* __restrict__ dst, const int* __restrict__ src,
                           const float* __restrict__ et, const float* __restrict__ nt,
                           const int* __restrict__ nid, const float* __restrict__ delta,
                           const unsigned* __restrict__ smax,
                           float* den, float* pos, int do_pos, int E) {
    int e = blockIdx.x * blockDim.x + threadIdx.x;
    if (e >= E) return;
    int d = dst[e];
    float td = nt[d] - et[e];
    float s = -delta[0] * td;
    float ee = expf(s - ordVal(smax[d]));
    atomicAdd(&den[d], ee);
    if (do_pos) {
        float p = (float)nid[src[e]] * expf(fabsf(td));
        atomicAdd(&pos[d], p);
    }
}

// ---- message gather + scatter-add: m[dst] += (zw*kappa) * x[src] ----
// 16 threads per edge; each thread moves 4 contiguous floats (float4 gather).
__global__ void edge_msg_k(const int* __restrict__ dst, const int* __restrict__ src,
                           const float* __restrict__ ew, const float* __restrict__ et,
                           const float* __restrict__ nt, const float* __restrict__ delta,
                           const float* __restrict__ ewg, const float* __restrict__ ewb,
                           const float* __restrict__ stats, const unsigned* __restrict__ smax,
                           const float* __restrict__ den, const float* __restrict__ xin,
                           float* m, int E) {
    long t = blockIdx.x * (long)blockDim.x + threadIdx.x;
    int e = (int)(t >> 4);
    if (e >= E) return;
    int sub = (int)(t & 15);
    int d = dst[e], sidx = src[e];
    float td = nt[d] - et[e];
    float s = -delta[0] * td;
    float ee = expf(s - ordVal(smax[d]));
    float kappa = ee / den[d];
    float mean = stats[2], var = stats[3];
    float zw = (ew[e] - mean) * rsqrtf(var + EPS) * ewg[0] + ewb[0];
    float coef = zw * kappa;
    const float4 xv = *(const float4*)(xin + (long)sidx * D + sub * 4);
    float* mp = m + (long)d * D + sub * 4;
    atomicAdd(mp + 0, coef * xv.x);
    atomicAdd(mp + 1, coef * xv.y);
    atomicAdd(mp + 2, coef * xv.z);
    atomicAdd(mp + 3, coef * xv.w);
}

// ---- combine the two linear layers: m@ws + m@wh == m@(ws+wh) ----
__global__ void combine_w_k(const float* ws, const float* wh, const float* bs,
                            const float* bh, float* wsum, float* bsum) {
    int i = blockIdx.x * blockDim.x + threadIdx.x;
    if (i < D * D) wsum[i] = ws[i] + wh[i];
    if (i < D)     bsum[i] = bs[i] + bh[i];
}

// ---- WMMA GEMM: C[M,64] = A[M,64] @ B[64,64] + bias ----
// One wave32 per 16x16 output tile; 4 waves/block cover the 64 columns of a
// 16-row strip. Uses V_WMMA_F32_16X16X4_F32 (full f32, matches reference).
__global__ void __launch_bounds__(128)
gemm_bias_k(const float* __restrict__ A, const float* __restrict__ B,
            const float* __restrict__ bias, float* __restrict__ C, int M) {
    int wave  = threadIdx.x >> 5;
    int lane  = threadIdx.x & 31;
    int mbase = blockIdx.x * 16;
    int nbase = wave * 16;
    if (mbase >= M) return;
    int lo = lane & 15, hi = lane >> 4;
    const float* Arow = A + (long)(mbase + lo) * D;
    v8f c = {};
    #pragma unroll
    for (int k0 = 0; k0 < D; k0 += 4) {
        int kc = k0 + 2 * hi;
        v2f a, b;
        a.x = Arow[kc];
        a.y = Arow[kc + 1];
        b.x = B[kc * D + nbase + lo];
        b.y = B[(kc + 1) * D + nbase + lo];
        c = __builtin_amdgcn_wmma_f32_16x16x4_f32(false, a, false, b,
                                                  (short)0, c, false, false);
    }
    int col = nbase + lo;
    float bv = bias[col];
    #pragma unroll
    for (int r = 0; r < 8; ++r) {
        int row = mbase + r + 8 * hi;
        C[(long)row * D + col] = c[r] + bv;
    }
}

// ---- WMMA GEMM with BN+ReLU folded into A, optional residual add ----
// out = relu(A*scale+shift) @ B + bias (+ residual)
__global__ void __launch_bounds__(128)
gemm_bnrelu_k(const float* __restrict__ A, const float* __restrict__ B,
              const float* __restrict__ bias, const float* __restrict__ scale,
              const float* __restrict__ shift, const float* __restrict__ residual,
              float* __restrict__ Cout, int M) {
    int wave  = threadIdx.x >> 5;
    int lane  = threadIdx.x & 31;
    int mbase = blockIdx.x * 16;
    int nbase = wave * 16;
    if (mbase >= M) return;
    int lo = lane & 15, hi = lane >> 4;
    const float* Arow = A + (long)(mbase + lo) * D;
    v8f c = {};
    #pragma unroll
    for (int k0 = 0; k0 < D; k0 += 4) {
        int kc = k0 + 2 * hi;
        v2f a, b;
        a.x = fmaxf(0.f, Arow[kc]     * scale[kc]     + shift[kc]);
        a.y = fmaxf(0.f, Arow[kc + 1] * scale[kc + 1] + shift[kc + 1]);
        b.x = B[kc * D + nbase + lo];
        b.y = B[(kc + 1) * D + nbase + lo];
        c = __builtin_amdgcn_wmma_f32_16x16x4_f32(false, a, false, b,
                                                  (short)0, c, false, false);
    }
    int col = nbase + lo;
    float bv = bias[col];
    #pragma unroll
    for (int r = 0; r < 8; ++r) {
        int row = mbase + r + 8 * hi;
        float v = c[r] + bv;
        if (residual) v += residual[(long)row * D + col];
        Cout[(long)row * D + col] = v;
    }
}

// ---- per-column stats over h (row-major, D=64 columns) ----
__global__ void colstats_k(const float* __restrict__ h, float* cs, long total) {
    long t = blockIdx.x * (long)blockDim.x + threadIdx.x;
    long stride = (long)gridDim.x * blockDim.x;   // multiple of 64
    int c = (int)(t & 63);
    float s = 0.f, q = 0.f;
    for (long i = t; i < total; i += stride) { float v = h[i]; s += v; q += v * v; }
    atomicAdd(&cs[c], s);
    atomicAdd(&cs[64 + c], q);
}
__global__ void bn_final_k(const float* cs, const float* bng, const float* bnb,
                           float* scale, float* shift, int Mrows) {
    int c = threadIdx.x;
    if (c < D) {
        float mean = cs[c] / (float)Mrows;
        float var  = cs[64 + c] / (float)Mrows - mean * mean;
        float s = bng[c] * rsqrtf(var + EPS);
        scale[c] = s;
        shift[c] = bnb[c] - mean * s;
    }
}

extern "C" void kernel_launch(void* const* d_in, const int* in_sizes, int n_in,
                              void* d_out, int out_size, void* d_ws, size_t ws_size,
                              hipStream_t stream) {
    (void)n_in; (void)out_size; (void)ws_size;
    const float* x   = (const float*)d_in[0];
    const int*   eix = (const int*)d_in[1];
    const float* ew  = (const float*)d_in[2];
    const float* et  = (const float*)d_in[3];
    const float* nt  = (const float*)d_in[4];
    const int*   nid = (const int*)d_in[5];

    const int N = in_sizes[0] / D;
    const int E = in_sizes[2];
    const int* dst = eix;        // edge_index[0] (targets)
    const int* src = eix + E;    // edge_index[1] (sources)

    // workspace layout (floats)
    float* wsf = (float*)d_ws;
    long off = 0;
    unsigned* smax = (unsigned*)(wsf + off); off += N;
    float* den   = wsf + off; off += N;
    float* m     = wsf + off; off += (long)N * D;
    float* h     = wsf + off; off += (long)N * D;
    float* x2    = wsf + off; off += (long)N * D;
    float* wsum  = wsf + off; off += D * D;
    float* bsum  = wsf + off; off += D;
    float* stats = wsf + off; off += 4;     // [sum, sumsq, mean, var] of edge weights
    float* cs    = wsf + off; off += 128;   // column [sum|sumsq]
    float* scale = wsf + off; off += D;
    float* shift = wsf + off; off += D;

    float* outp = (float*)d_out;
    float* posp = outp + (long)N * D;

    // edge-weight batch stats (weights identical for both layers)
    fill_f32<<<1, 256, 0, stream>>>(stats, 0.f, 4);
    ew_stats_k<<<256, 256, 0, stream>>>(ew, E, stats);
    ew_final_k<<<1, 1, 0, stream>>>(stats, E);

    const int eblocks   = (E + 255) / 256;
    const long msgT     = (long)E * 16;
    const int msgblocks = (int)((msgT + 255) / 256);
    const int mtiles    = (N + 15) / 16;       // N = 100000 -> 6250 exact
    const int ndblocks  = (int)(((long)N * D + 255) / 256);

    for (int L = 0; L < 2; ++L) {
        const int base = (L == 0) ? 6 : 17;
        const float* delta = (const float*)d_in[base + 0];
        const float* ewg   = (const float*)d_in[base + 1];
        const float* ewb   = (const float*)d_in[base + 2];
        const float* Ws    = (const float*)d_in[base + 3];
        const float* Bs    = (const float*)d_in[base + 4];
        const float* Wh    = (const float*)d_in[base + 5];
        const float* Bh    = (const float*)d_in[base + 6];
        const float* Bng   = (const float*)d_in[base + 7];
        const float* Bnb   = (const float*)d_in[base + 8];
        const float* Fw    = (const float*)d_in[base + 9];
        const float* Fb    = (const float*)d_in[base + 10];
        const float* xin   = (L == 0) ? x : x2;

        fill_f32<<<(N + 255) / 256, 256, 0, stream>>>((float*)smax, 0.f, N); // 0 == ord(-inf) floor
        fill_f32<<<(N + 255) / 256, 256, 0, stream>>>(den, 0.f, N);
        fill_f32<<<ndblocks, 256, 0, stream>>>(m, 0.f, (long)N * D);
        fill_f32<<<1, 128, 0, stream>>>(cs, 0.f, 128);
        if (L == 1) fill_f32<<<(N + 255) / 256, 256, 0, stream>>>(posp, 0.f, N);

        edge_max_k<<<eblocks, 256, 0, stream>>>(dst, et, nt, delta, smax, E);
        edge_den_k<<<eblocks, 256, 0, stream>>>(dst, src, et, nt, nid, delta, smax,
                                                den, posp, (L == 1) ? 1 : 0, E);
        edge_msg_k<<<msgblocks, 256, 0, stream>>>(dst, src, ew, et, nt, delta, ewg, ewb,
                                                  stats, smax, den, xin, m, E);

        combine_w_k<<<(D * D + 255) / 256, 256, 0, stream>>>(Ws, Wh, Bs, Bh, wsum, bsum);
        gemm_bias_k<<<mtiles, 128, 0, stream>>>(m, wsum, bsum, h, N);
        colstats_k<<<512, 256, 0, stream>>>(h, cs, (long)N * D);
        bn_final_k<<<1, 64, 0, stream>>>(cs, Bng, Bnb, scale, shift, N);

        if (L == 0) {
            gemm_bnrelu_k<<<mtiles, 128, 0, stream>>>(h, Fw, Fb, scale, shift, x, x2, N);
        } else {
            gemm_bnrelu_k<<<mtiles, 128, 0, stream>>>(h, Fw, Fb, scale, shift, nullptr, outp, N);
        }
    }
}